// PARFLM_90254442758430
// MI455X (gfx1250) — compile-verified
//
#include <hip/hip_runtime.h>
#include <math.h>

typedef float v2f __attribute__((ext_vector_type(2)));
typedef float v8f __attribute__((ext_vector_type(8)));

#define B_   8
#define T_   512
#define D_   128
#define DL_  16
#define K_   8
#define HP_  32
#define HT_  32
#define BT_  (B_*T_)
#define EPS2_ 1.0e-4f

__device__ __forceinline__ float gelu_exact(float x) {
    // torch F.gelu default (erf form)
    return 0.5f * x * (1.0f + erff(x * 0.70710678118654752f));
}
__device__ __forceinline__ float softplus_(float x) {
    return (x > 20.0f) ? x : log1pf(expf(x));
}

// ---------------------------------------------------------------------------
// Phase 1: per (b,t) row projections into workspace.
//  ws layout (floats):
//   projT [BT*32] | projU [BT*32] | lq [BT*16] | ls [BT*16] |
//   n2h [BT] | n2hs [BT] | n2lq [BT] | n2ls [BT]      (total BT*100 floats)
// ---------------------------------------------------------------------------
__global__ void __launch_bounds__(256) proj_kernel(
    const float* __restrict__ h,  const float* __restrict__ hsrc,
    const float* __restrict__ Wl, const float* __restrict__ Wt,
    const float* __restrict__ twq, const float* __restrict__ tws,
    const float* __restrict__ twd, const float* __restrict__ tb1,
    float* __restrict__ ws)
{
    int idx = blockIdx.x * blockDim.x + threadIdx.x;   // over B*T
    if (idx >= BT_) return;
    const float* hr = h    + (size_t)idx * D_;
    const float* sr = hsrc + (size_t)idx * D_;

    float lq[DL_], lsv[DL_], thq[K_], ths[K_];
    #pragma unroll
    for (int i = 0; i < DL_; ++i) { lq[i] = 0.f; lsv[i] = 0.f; }
    #pragma unroll
    for (int i = 0; i < K_; ++i)  { thq[i] = 0.f; ths[i] = 0.f; }
    float n2q = 0.f, n2s = 0.f;

    for (int d = 0; d < D_; ++d) {
        float a = hr[d], b = sr[d];
        n2q = fmaf(a, a, n2q);
        n2s = fmaf(b, b, n2s);
        #pragma unroll
        for (int l = 0; l < DL_; ++l) {
            float w = Wl[l * D_ + d];
            lq[l]  = fmaf(a, w, lq[l]);
            lsv[l] = fmaf(b, w, lsv[l]);
        }
        #pragma unroll
        for (int k = 0; k < K_; ++k) {
            float w = Wt[k * D_ + d];
            thq[k] = fmaf(a, w, thq[k]);
            ths[k] = fmaf(b, w, ths[k]);
        }
    }
    float n2lq = 0.f, n2ls = 0.f;
    #pragma unroll
    for (int l = 0; l < DL_; ++l) {
        n2lq = fmaf(lq[l],  lq[l],  n2lq);
        n2ls = fmaf(lsv[l], lsv[l], n2ls);
    }

    float* projT = ws;
    float* projU = ws + (size_t)BT_ * 32;
    float* lqO   = ws + (size_t)BT_ * 64;
    float* lsO   = ws + (size_t)BT_ * 80;
    float* n2hO  = ws + (size_t)BT_ * 96;
    float* n2hsO = n2hO  + BT_;
    float* n2lqO = n2hsO + BT_;
    float* n2lsO = n2lqO + BT_;

    #pragma unroll
    for (int l = 0; l < DL_; ++l) {
        lqO[idx * DL_ + l] = lq[l];
        lsO[idx * DL_ + l] = lsv[l];
    }
    n2hO[idx] = n2q; n2hsO[idx] = n2s; n2lqO[idx] = n2lq; n2lsO[idx] = n2ls;

    // proj_t = th_q @ (tw_q + tw_d)^T + b1 ; proj_u = th_s @ (tw_s - tw_d)^T
    for (int m = 0; m < HT_; ++m) {
        float st = tb1[m], su = 0.f;
        #pragma unroll
        for (int k = 0; k < K_; ++k) {
            st = fmaf(thq[k], twq[m * K_ + k] + twd[m * K_ + k], st);
            su = fmaf(ths[k], tws[m * K_ + k] - twd[m * K_ + k], su);
        }
        projT[(size_t)idx * 32 + m] = st;
        projU[(size_t)idx * 32 + m] = su;
    }
}

// ---------------------------------------------------------------------------
// Phase 2: one wave32 per 16x16 output tile. WMMA f32 16x16x4 accumulates
// h.h_src^T (K=128) and l_q.l_s^T (K=16); fused per-pair MLPs on registers.
// ---------------------------------------------------------------------------
__global__ void __launch_bounds__(256) pair_kernel(
    const float* __restrict__ h,      const float* __restrict__ hsrc,
    const float* __restrict__ phi_w1, const float* __restrict__ phi_b1,
    const float* __restrict__ phi_w2, const float* __restrict__ phi_b2,
    const float* __restrict__ tw2_w,  const float* __restrict__ tw2_b,
    const float* __restrict__ ws,     float* __restrict__ out)
{
    __shared__ float sW1[32], sB1[32], sW2[32], sWT[32];
    __shared__ float sPT[8][16][32];
    __shared__ float sPU[8][16][32];

    const int tid  = threadIdx.x;
    const int w    = tid >> 5;         // wave id in block (0..7)
    const int lane = tid & 31;
    const int row  = lane & 15;        // N (col) for B/C/D, M%16 for A
    const int half = lane >> 4;        // K-half for A/B, M-half for C/D

    const int tilesPerB = (T_/16) * (T_/16);       // 1024
    const int tile = blockIdx.x * 8 + w;
    const int b   = tile / tilesPerB;
    const int rem = tile % tilesPerB;
    const int t0  = (rem / (T_/16)) * 16;
    const int s0  = (rem % (T_/16)) * 16;

    const float* projT = ws;
    const float* projU = ws + (size_t)BT_ * 32;
    const float* lqP   = ws + (size_t)BT_ * 64;
    const float* lsP   = ws + (size_t)BT_ * 80;
    const float* n2h   = ws + (size_t)BT_ * 96;
    const float* n2hs  = n2h  + BT_;
    const float* n2lq  = n2hs + BT_;
    const float* n2ls  = n2lq + BT_;

    if (tid < 32) {
        sW1[tid] = phi_w1[tid];
        sB1[tid] = phi_b1[tid];
        sW2[tid] = phi_w2[tid];
        sWT[tid] = tw2_w[tid];
    }
    // stage this wave's proj_t / proj_u 16x32 tiles (no divergence: 16 iters each)
    for (int idx = lane; idx < 512; idx += 32) {
        int m = idx >> 5, j = idx & 31;
        sPT[w][m][j] = projT[((size_t)(b * T_ + t0 + m)) * 32 + j];
        sPU[w][m][j] = projU[((size_t)(b * T_ + s0 + m)) * 32 + j];
    }
    __syncthreads();

    // ---- WMMA: A 16x4 layout => lane(row,half) holds K = 2*half + {0,1} ----
    v8f acc  = {};   // h . h_src^T  (K = 128)
    v8f accL = {};   // l_q . l_s^T  (K = 16)
    const float* pA = h    + (size_t)(b * T_ + t0 + row) * D_ + 2 * half;
    const float* pB = hsrc + (size_t)(b * T_ + s0 + row) * D_ + 2 * half;
    #pragma unroll 4
    for (int k = 0; k < D_; k += 4) {
        v2f av = *(const v2f*)(pA + k);
        v2f bv = *(const v2f*)(pB + k);
        acc = __builtin_amdgcn_wmma_f32_16x16x4_f32(false, av, false, bv,
                                                    (short)0, acc, false, false);
    }
    const float* pLA = lqP + (size_t)(b * T_ + t0 + row) * DL_ + 2 * half;
    const float* pLB = lsP + (size_t)(b * T_ + s0 + row) * DL_ + 2 * half;
    #pragma unroll
    for (int k = 0; k < DL_; k += 4) {
        v2f av = *(const v2f*)(pLA + k);
        v2f bv = *(const v2f*)(pLB + k);
        accL = __builtin_amdgcn_wmma_f32_16x16x4_f32(false, av, false, bv,
                                                     (short)0, accL, false, false);
    }

    // ---- per-pair math on the C-layout registers ----
    const int s = s0 + row;                       // N = row for C/D
    const float n2s_s  = n2hs[b * T_ + s];
    const float n2ls_s = n2ls[b * T_ + s];

    float ld2[8], rinv[8], hsum[8], tsum[8];
    #pragma unroll
    for (int i = 0; i < 8; ++i) {                 // M = i + 8*half
        int t = t0 + i + 8 * half;
        float d2l = fmaxf(n2lq[b * T_ + t] + n2ls_s - 2.0f * accL[i], 0.0f);
        float d2  = fmaxf(n2h[b * T_ + t]  + n2s_s  - 2.0f * acc[i],  0.0f);
        ld2[i]  = d2l;
        rinv[i] = rsqrtf(d2 + EPS2_);             // 1/sqrt(dist2 + eps^2)
        hsum[i] = 0.0f;
        tsum[i] = 0.0f;
    }

    for (int j = 0; j < 32; ++j) {
        float w1 = sW1[j], b1 = sB1[j], w2 = sW2[j], wt = sWT[j];
        float pu = sPU[w][row][j];
        #pragma unroll
        for (int i = 0; i < 8; ++i) {
            float x = fmaf(ld2[i], w1, b1);                  // Phi hidden
            hsum[i] = fmaf(gelu_exact(x), w2, hsum[i]);
            float pt = sPT[w][i + 8 * half][j];              // Theta hidden
            tsum[i] = fmaf(gelu_exact(pt + pu), wt, tsum[i]);
        }
    }

    const float b2  = phi_b2[0];
    const float tb2 = tw2_b[0];
    #pragma unroll
    for (int i = 0; i < 8; ++i) {
        int t = t0 + i + 8 * half;
        float c     = softplus_(hsum[i] + b2);
        float Phi   = expf(-c * ld2[i]);
        float Theta = tanhf(tsum[i] + tb2);
        out[(size_t)b * T_ * T_ + (size_t)t * T_ + s] = -Theta * Phi * rinv[i];
    }
}

extern "C" void kernel_launch(void* const* d_in, const int* in_sizes, int n_in,
                              void* d_out, int out_size, void* d_ws, size_t ws_size,
                              hipStream_t stream) {
    (void)in_sizes; (void)n_in; (void)out_size; (void)ws_size;
    const float* h      = (const float*)d_in[0];
    const float* hsrc   = (const float*)d_in[1];
    const float* Wl     = (const float*)d_in[2];
    const float* Wt     = (const float*)d_in[3];
    const float* phi_w1 = (const float*)d_in[4];
    const float* phi_b1 = (const float*)d_in[5];
    const float* phi_w2 = (const float*)d_in[6];
    const float* phi_b2 = (const float*)d_in[7];
    const float* twq    = (const float*)d_in[8];
    const float* tws    = (const float*)d_in[9];
    const float* twd    = (const float*)d_in[10];
    const float* tb1    = (const float*)d_in[11];
    const float* tw2_w  = (const float*)d_in[12];
    const float* tw2_b  = (const float*)d_in[13];
    float* ws  = (float*)d_ws;
    float* out = (float*)d_out;

    proj_kernel<<<(BT_ + 255) / 256, 256, 0, stream>>>(
        h, hsrc, Wl, Wt, twq, tws, twd, tb1, ws);

    const int tiles = B_ * (T_/16) * (T_/16);     // 8192 tiles, 8 waves/block
    pair_kernel<<<tiles / 8, 256, 0, stream>>>(
        h, hsrc, phi_w1, phi_b1, phi_w2, phi_b2, tw2_w, tw2_b, ws, out);
}